// VehicleSideFCOSTarget_82884278879268
// MI455X (gfx1250) — compile-verified
//
#include <hip/hip_runtime.h>
#include <stdint.h>

#define INFV        100000000.0f
#define NUM_GTS     512
#define BACKGROUND  80
#define BLOCK       256

typedef unsigned int u32x4 __attribute__((ext_vector_type(4)));
typedef int          i32x8 __attribute__((ext_vector_type(8)));
typedef int          i32x4 __attribute__((ext_vector_type(4)));

__global__ __launch_bounds__(BLOCK) void fcos_target_kernel(
    const float* __restrict__ points,        // (N,2)
    const float* __restrict__ regress_ranges,// (N,2)
    const float* __restrict__ strides_pts,   // (N,)  already s*1.5
    const float* __restrict__ gt_bboxes,     // (512,4)
    const int*   __restrict__ gt_labels,     // (512,)
    float*       __restrict__ out,           // labels(N) | bt(4N) | weights(N)
    int N)
{
    __shared__ float4 s_box[NUM_GTS];   // DMA destination: x1,y1,x2,y2 interleaved
    __shared__ float4 s_aux[NUM_GTS];   // cx, cy, area, pad

    // ---- Stage the 8KB GT table into LDS with one TDM descriptor (wave 0) ----
    if (threadIdx.x == 0) {
        uint64_t ga  = (uint64_t)(uintptr_t)gt_bboxes;
        uint32_t lds = (uint32_t)(uintptr_t)(&s_box[0]);   // low 32 bits = LDS offset

        // D# group 0 (128b): [1:0]=count=1, [63:32]=lds_addr,
        // [120:64]=global_addr, [127:126]=type=2
        u32x4 g0;
        g0[0] = 1u;
        g0[1] = lds;
        g0[2] = (uint32_t)ga;
        g0[3] = (uint32_t)(ga >> 32) | 0x80000000u;

        // D# group 1 (256b): data_size=4B, 1-D tile of 2048 dwords
        i32x8 g1;
        g1[0] = 0x00020000;          // workgroup_mask=0, data_size=4B
        g1[1] = 0x08000000;          // tensor_dim0[15:0]=2048 in bits[63:48]
        g1[2] = 0x00010000;          // tensor_dim1=1
        g1[3] = 0x08000000;          // tile_dim0=2048
        g1[4] = 0x00000001;          // tile_dim1=1, tile_dim2=0
        g1[5] = 2048;                // tensor_dim0_stride[31:0]
        g1[6] = 0;
        g1[7] = 0;

        i32x4 gz4 = {0, 0, 0, 0};    // groups 2/3 unused (tile_dim3/4 = 0)
        i32x8 gz8 = {0, 0, 0, 0, 0, 0, 0, 0};

        __builtin_amdgcn_tensor_load_to_lds(g0, g1, gz4, gz4, gz8, 0);
        __builtin_amdgcn_s_wait_tensorcnt(0);
    }
    __syncthreads();

    // ---- Precompute per-GT invariants into LDS (packed for single b128 load) ----
    for (int j = threadIdx.x; j < NUM_GTS; j += BLOCK) {
        float4 b = s_box[j];
        float4 a;
        a.x = (b.x + b.z) * 0.5f;          // cx
        a.y = (b.y + b.w) * 0.5f;          // cy
        a.z = (b.z - b.x) * (b.w - b.y);   // area
        a.w = 0.0f;
        s_aux[j] = a;
    }
    __syncthreads();

    int i = blockIdx.x * BLOCK + threadIdx.x;
    if (i >= N) return;

    const float2 pt = ((const float2*)points)[i];
    const float2 rg = ((const float2*)regress_ranges)[i];
    const float  x = pt.x, y = pt.y;
    const float  lo = rg.x, hi = rg.y;
    const float  rad = strides_pts[i];

    float best = 3.0e38f;   // > INFV so the first GT always wins a tie (argmin==0 when all masked)
    int   bidx = 0;

#pragma unroll 4
    for (int j = 0; j < NUM_GTS; ++j) {
        float4 b = s_box[j];             // LDS broadcast: same j across the wave
        float4 a = s_aux[j];

        float l   = x - b.x;
        float t   = y - b.y;
        float r   = b.z - x;
        float bo  = b.w - y;
        float mxd = fmaxf(fmaxf(l, t), fmaxf(r, bo));   // -> v_max3 chain
        float mnd = fminf(fminf(l, t), fminf(r, bo));   // -> v_min3 chain

        // min(x-x0, y-y0, x1-x, y1-y)  ==  min(mnd, rad-|cx-x|, rad-|cy-y|)
        float cx_d = rad - __builtin_fabsf(a.x - x);    // abs as src modifier
        float cy_d = rad - __builtin_fabsf(a.y - y);
        float cmn  = fminf(mnd, fminf(cx_d, cy_d));

        bool ok = (cmn > 0.0f) && (mxd >= lo) && (mxd <= hi);
        float av = ok ? a.z : INFV;
        if (av < best) { best = av; bidx = j; }
    }

    // Winning box targets + label
    float4 wb = s_box[bidx];
    out[i] = (best == INFV) ? (float)BACKGROUND : (float)gt_labels[bidx];

    float* bt = out + N;
    bt[4 * i + 0] = x - wb.x;
    bt[4 * i + 1] = y - wb.y;
    bt[4 * i + 2] = wb.z - x;
    bt[4 * i + 3] = wb.w - y;

    out[5 * N + i] = 1.0f;
}

extern "C" void kernel_launch(void* const* d_in, const int* in_sizes, int n_in,
                              void* d_out, int out_size, void* d_ws, size_t ws_size,
                              hipStream_t stream) {
    const float* points   = (const float*)d_in[0];
    const float* rr       = (const float*)d_in[1];
    const float* strides  = (const float*)d_in[2];
    const float* gt_boxes = (const float*)d_in[3];
    const int*   gt_lab   = (const int*)d_in[4];
    float* out = (float*)d_out;

    int N = in_sizes[2];                 // strides_pts has one entry per point
    int grid = (N + BLOCK - 1) / BLOCK;

    fcos_target_kernel<<<grid, BLOCK, 0, stream>>>(points, rr, strides, gt_boxes,
                                                   gt_lab, out, N);
}